// MS_SSIM_L1_LOSS_76450417868805
// MI455X (gfx1250) — compile-verified
//
#include <hip/hip_runtime.h>
#include <hip/hip_bf16.h>

#if defined(__HIP_DEVICE_COMPILE__)
#if !__has_builtin(__builtin_amdgcn_wmma_f32_16x16x4_f32)
#error "gfx1250 f32 WMMA builtin (__builtin_amdgcn_wmma_f32_16x16x4_f32) not available"
#endif
#endif

typedef float v2f __attribute__((ext_vector_type(2)));
typedef float v8f __attribute__((ext_vector_type(8)));

namespace {
constexpr int   TILE  = 16;   // output tile edge
constexpr int   HALO  = 16;   // PAD = 2*sigma_max
constexpr int   REGN  = 48;   // TILE + 2*HALO
constexpr int   IN_W  = 49;   // padded LDS stride (odd -> conflict-free column walks)
constexpr int   TMP_W = 17;   // padded LDS stride for H-pass result
constexpr int   FS    = 33;   // filter taps
constexpr int   IMG   = 512;
constexpr long  PLANE = 8L * 512 * 512;           // elements per ws plane
constexpr float C1v   = 0.0001f;                  // (0.01*DR)^2
constexpr float C2v   = 0.0009f;                  // (0.03*DR)^2

// slot 0..6: distinct (sigma, channel) pairs of the grouped conv
// slot 7..9: sigma=8 blur of |x-y| for channel slot-7 (gaussian_l1 path)
constexpr int kSigma[10] = {0, 1, 1, 2, 3, 3, 4, 4, 4, 4};
constexpr int kChan[10]  = {0, 0, 1, 1, 1, 2, 2, 0, 1, 2};
} // namespace

// D = A(16x4,f32) * B(4x16,f32) + C(16x16,f32) on the CDNA5 matrix unit.
__device__ __forceinline__ v8f wmma4(v2f a, v2f b, v8f c) {
#if defined(__HIP_DEVICE_COMPILE__)
  return __builtin_amdgcn_wmma_f32_16x16x4_f32(false, a, false, b, (short)0, c,
                                               false, false);
#else
  return c;
#endif
}

// Banded weight: g[k] for k in [0,33), else 0.
__device__ __forceinline__ float bandw(const float* g, int k) {
  return (k >= 0 && k < FS) ? g[k] : 0.0f;
}

// Quantity value at LDS tile coords (r,c). x,y are denormalized (0.5*v+0.5).
// q: 0=x 1=y 2=x^2 3=y^2 4=x*y 5=|x-y|
__device__ __forceinline__ float qval(const float* xs, const float* ys,
                                      int r, int c, int q) {
  const float xv = xs[r * IN_W + c] * 0.5f + 0.5f;
  const float yv = ys[r * IN_W + c] * 0.5f + 0.5f;
  switch (q) {
    case 0:  return xv;
    case 1:  return yv;
    case 2:  return xv * xv;
    case 3:  return yv * yv;
    case 4:  return xv * yv;
    default: return fabsf(xv - yv);
  }
}

// One separable 33-tap blur of a 16x16 tile, both passes as banded K=4 f32 WMMA
// GEMMs.  wf holds the per-lane band fragments (shared by H-pass B and V-pass A).
__device__ __forceinline__ v8f blur_one(int q, const float* xin, const float* yin,
                                        float* tmp, const v2f* wf, int lane) {
  const int m  = lane & 15;
  const int kl = (lane >> 4) << 1;      // 0 or 2 (K lane split)
  __syncthreads();                      // tmp reuse fence across quantities

  // ---- Horizontal pass: tmp(48x16) = Q(48x48) * Wband(48x16), 3 row-tiles ----
  v8f a0 = {}, a1 = {}, a2 = {};
#pragma unroll
  for (int kc = 0; kc < 12; ++kc) {
    const int c = kc * 4 + kl;
    const v2f bw = wf[kc];
    v2f a;
    a.x = qval(xin, yin, m, c, q);       a.y = qval(xin, yin, m, c + 1, q);
    a0 = wmma4(a, bw, a0);
    a.x = qval(xin, yin, 16 + m, c, q);  a.y = qval(xin, yin, 16 + m, c + 1, q);
    a1 = wmma4(a, bw, a1);
    a.x = qval(xin, yin, 32 + m, c, q);  a.y = qval(xin, yin, 32 + m, c + 1, q);
    a2 = wmma4(a, bw, a2);
  }
  // D layout: VGPR v -> row v (lanes 0-15) / row v+8 (lanes 16-31), col = lane&15.
  const int half = (lane >> 4) * 8;
#pragma unroll
  for (int v = 0; v < 8; ++v) {
    tmp[(half + v) * TMP_W + m]      = a0[v];
    tmp[(16 + half + v) * TMP_W + m] = a1[v];
    tmp[(32 + half + v) * TMP_W + m] = a2[v];
  }
  __syncthreads();

  // ---- Vertical pass: out(16x16) = Wband^T(16x48) * tmp(48x16) ----
  v8f d = {};
#pragma unroll
  for (int kc = 0; kc < 12; ++kc) {
    const int c = kc * 4 + kl;
    v2f b;
    b.x = tmp[c * TMP_W + m];
    b.y = tmp[(c + 1) * TMP_W + m];
    d = wmma4(wf[kc], b, d);
  }
  return d;
}

__global__ void __launch_bounds__(32)
msl1_blur_kernel(const float* __restrict__ X, const float* __restrict__ Y,
                 const float* __restrict__ GM, float* __restrict__ planes) {
  __shared__ float g1d[FS];
  __shared__ float xin[REGN * IN_W];
  __shared__ float yin[REGN * IN_W];
  __shared__ float tmp[REGN * TMP_W];

  const int lane = threadIdx.x;          // one wave32 per block
  const int tx = blockIdx.x & 31;
  const int ty = blockIdx.x >> 5;
  const int b = blockIdx.y;
  const int slot = blockIdx.z;

  const int sigma = kSigma[slot];
  const int chan  = kChan[slot];

  // Recover normalized 1D gaussian from 2D mask center row: g[i]=K[16][i]/sqrt(K[16][16])
  {
    const float* krow = GM + (long)(3 * sigma) * FS * FS + 16 * FS;
    for (int i = lane; i < FS; i += 32) g1d[i] = krow[i];
  }
  __syncthreads();
  const float nrm = 1.0f / sqrtf(g1d[16]);
  __syncthreads();
  for (int i = lane; i < FS; i += 32) g1d[i] *= nrm;
  __syncthreads();

  // Stage the 48x48 halo'd input region (zero padding outside the image).
  const long pbase = (long)(b * 3 + chan) * IMG * IMG;
  const int r0 = ty * TILE - HALO;
  const int c0 = tx * TILE - HALO;
  __builtin_prefetch(&X[pbase + (long)(ty * TILE) * IMG], 0, 1);  // global_prefetch_b8
  __builtin_prefetch(&Y[pbase + (long)(ty * TILE) * IMG], 0, 1);
  for (int i = lane; i < REGN * REGN; i += 32) {
    const int rr = i / REGN, cc = i - rr * REGN;
    const int gr = r0 + rr, gc = c0 + cc;
    const bool ok = (gr >= 0) && (gr < IMG) && (gc >= 0) && (gc < IMG);
    const long idx = pbase + (long)gr * IMG + gc;
    xin[rr * IN_W + cc] = ok ? X[idx] : 0.0f;
    yin[rr * IN_W + cc] = ok ? Y[idx] : 0.0f;
  }
  __syncthreads();

  // Band fragments, resident in registers for both passes and all quantities.
  // A-frag(16x4): a.x=A[m][kl], a.y=A[m][kl+1]; B-frag(4x16): b.x=B[kl][n], b.y=B[kl+1][n]
  // -> for our band matrices both reduce to band(kc*4+kl+{0,1}-(lane&15)).
  const int m  = lane & 15;
  const int kl = (lane >> 4) << 1;
  v2f wf[12];
#pragma unroll
  for (int kc = 0; kc < 12; ++kc) {
    const int c = kc * 4 + kl;
    wf[kc].x = bandw(g1d, c - m);
    wf[kc].y = bandw(g1d, c + 1 - m);
  }

  v8f outq[5];
  if (slot < 7) {
#pragma unroll
    for (int q = 0; q < 5; ++q) outq[q] = blur_one(q, xin, yin, tmp, wf, lane);
  } else {
    outq[0] = blur_one(5, xin, yin, tmp, wf, lane);
  }

  // Fused SSIM algebra in registers; only per-pair cs/l (and L1 blur) hit memory.
  const int half = (lane >> 4) * 8;
  const int col = tx * TILE + m;
#pragma unroll
  for (int v = 0; v < 8; ++v) {
    const int row = ty * TILE + half + v;
    const long ob = (long)b * (IMG * IMG) + (long)row * IMG + col;
    if (slot < 7) {
      const float mux = outq[0][v], muy = outq[1][v];
      const float mxx = outq[2][v], myy = outq[3][v], mxy = outq[4][v];
      const float mux2 = mux * mux, muy2 = muy * muy;
      const float l  = (2.0f * mux * muy + C1v) / (mux2 + muy2 + C1v);
      const float cs = (2.0f * (mxy - mux * muy) + C2v) /
                       ((mxx - mux2) + (myy - muy2) + C2v);
      planes[(long)slot * PLANE + ob] = cs;
      if (slot == 6) planes[7 * PLANE + ob] = l;
    } else {
      planes[(long)(8 + slot - 7) * PLANE + ob] = outq[0][v];
    }
  }
}

__device__ __forceinline__ void waveReduceAdd(float v, float* dst) {
#pragma unroll
  for (int o = 16; o > 0; o >>= 1) v += __shfl_down(v, o, 32);
  if ((threadIdx.x & 31) == 0) atomicAdd(dst, v);
}

__global__ void __launch_bounds__(256)
msl1_combine_kernel(const float* __restrict__ planes, float* __restrict__ acc) {
  const long i = (long)blockIdx.x * 256 + threadIdx.x;   // exactly PLANE threads
  const float cs0 = planes[0 * PLANE + i];
  const float cs1 = planes[1 * PLANE + i];
  const float cs2 = planes[2 * PLANE + i];
  const float cs3 = planes[3 * PLANE + i];
  const float cs4 = planes[4 * PLANE + i];
  const float cs5 = planes[5 * PLANE + i];
  const float cs6 = planes[6 * PLANE + i];
  const float l6  = planes[7 * PLANE + i];
  // PIcs with pair multiplicities 3,2,1,3,1,2,3 ; lM = l_pair6^3
  float pics = cs0 * cs0 * cs0;
  pics *= cs1 * cs1;
  pics *= cs2;
  pics *= cs3 * cs3 * cs3;
  pics *= cs4;
  pics *= cs5 * cs5;
  pics *= cs6 * cs6 * cs6;
  const float lm  = l6 * l6 * l6;
  const float gl1 = (planes[8 * PLANE + i] + planes[9 * PLANE + i] +
                     planes[10 * PLANE + i]) * (1.0f / 3.0f);
  const float mix = 200.0f * (0.025f * (1.0f - lm * pics) + 0.975f * gl1);
  waveReduceAdd(mix, &acc[0]);
}

__global__ void __launch_bounds__(256)
msl1_l1_kernel(const float* __restrict__ X, const float* __restrict__ Y,
               float* __restrict__ acc) {
  const long i = (long)blockIdx.x * 256 + threadIdx.x;   // exactly 8*3*512*512
  waveReduceAdd(0.5f * fabsf(X[i] - Y[i]), &acc[1]);     // |denorm(x)-denorm(y)|
}

__global__ void __launch_bounds__(256)
msl1_disc_kernel(const float* __restrict__ D, float* __restrict__ acc) {
  const long i = (long)blockIdx.x * 256 + threadIdx.x;
  float v = 0.0f;
  if (i < 30752) { const float d = D[i] - 1.0f; v = d * d; }
  waveReduceAdd(v, &acc[2]);
}

__global__ void msl1_init_kernel(float* acc) {
  if (threadIdx.x < 16) acc[threadIdx.x] = 0.0f;
}

__global__ void msl1_final_kernel(const float* __restrict__ acc,
                                  float* __restrict__ out) {
  const float meanMix = acc[0] * (1.0f / 2097152.0f);   // 8*512*512
  const float meanL1  = acc[1] * (1.0f / 6291456.0f);   // 8*3*512*512
  const float meanMse = acc[2] * (1.0f / 30752.0f);     // 8*62*62
  out[0] = 0.5f * (meanMix + 100.0f * meanL1 + meanMse);
}

extern "C" void kernel_launch(void* const* d_in, const int* in_sizes, int n_in,
                              void* d_out, int out_size, void* d_ws, size_t ws_size,
                              hipStream_t stream) {
  const float* X  = (const float*)d_in[0];   // (8,3,512,512)
  const float* Y  = (const float*)d_in[1];   // (8,3,512,512)
  const float* Dc = (const float*)d_in[2];   // (8,1,62,62)
  const float* GM = (const float*)d_in[3];   // (15,1,33,33)
  float* out = (float*)d_out;

  // ws layout: [0..15] scalar accumulators, then 11 planes of 8*512*512 f32
  // (7 cs + 1 l + 3 L1-blur) = ~92.3 MB, L2-resident.
  float* acc    = (float*)d_ws;
  float* planes = acc + 16;

  msl1_init_kernel<<<1, 32, 0, stream>>>(acc);
  msl1_blur_kernel<<<dim3(1024, 8, 10), 32, 0, stream>>>(X, Y, GM, planes);
  msl1_combine_kernel<<<8192, 256, 0, stream>>>(planes, acc);
  msl1_l1_kernel<<<24576, 256, 0, stream>>>(X, Y, acc);
  msl1_disc_kernel<<<121, 256, 0, stream>>>(Dc, acc);
  msl1_final_kernel<<<1, 1, 0, stream>>>(acc, out);
}